// voxel_3d_generator_56676388438716
// MI455X (gfx1250) — compile-verified
//
#include <hip/hip_runtime.h>

// ---------------------------------------------------------------------------
// Scatter-mean (voxelization) for MI455X / gfx1250.
//
// Roofline: ~150 MB of traffic, 0 matmul FLOPs -> pure data-movement op,
// ~6.5 us HBM floor. Accumulators (18 MB + 5.2 MB) live in the 192 MB L2, so
// the scatter phase is L2-atomic bound. CDNA5-specific paths used:
//   * global_load_async_to_lds_b128  (ASYNCcnt async DMA staging, no VGPR hop)
//   * s_wait_asynccnt                (split dependency counter)
//   * global_atomic_add_f32/u32      (native no-return L2 atomics via asm,
//                                     offset folded into the 24-bit IOFFSET;
//                                     STOREcnt drained by S_ENDPGM's implicit
//                                     S_WAIT_IDLE)
// WMMA is intentionally absent: the op has no matrix-multiply structure.
// ---------------------------------------------------------------------------

#define N_PTS  1048576
#define C_PT   9
#define C_LB   20
#define V1_    500000
#define V8_    65536

__device__ __forceinline__ void async_ld_b128(unsigned lds_off, unsigned long long gaddr) {
  // GV mode: dst VGPR = LDS byte offset, src VGPR pair = 64-bit global address.
  asm volatile("global_load_async_to_lds_b128 %0, %1, off"
               :: "v"(lds_off), "v"(gaddr) : "memory");
}

__device__ __forceinline__ void wait_async0() {
  asm volatile("s_wait_asynccnt 0" ::: "memory");
}

// Non-returning native atomics (fire-and-forget, tracked by STOREcnt only).
__device__ __forceinline__ void gatom_add_u32(unsigned long long addr, unsigned v) {
  asm volatile("global_atomic_add_u32 %0, %1, off" :: "v"(addr), "v"(v));
}

// Unroll a row of C consecutive f32 atomic adds with compile-time immediate
// offsets (template recursion so the "i" constraint is a true constant).
template <int C, int I = 0>
struct RowAtomAdd {
  static __device__ __forceinline__ void run(unsigned long long base,
                                             const float* __restrict__ row) {
    asm volatile("global_atomic_add_f32 %0, %1, off offset:%c2"
                 :: "v"(base), "v"(row[I]), "i"(I * 4));
    RowAtomAdd<C, I + 1>::run(base, row);
  }
};
template <int C>
struct RowAtomAdd<C, C> {
  static __device__ __forceinline__ void run(unsigned long long, const float*) {}
};

// ---- labels: 256-point tile, 256 threads, 20 KB LDS -----------------------
// (placed first in the file so the asm snippet shows the atomic section)
__global__ __launch_bounds__(256)
void scatter_labels_kernel(const float* __restrict__ labels,
                           const int*   __restrict__ idx8,
                           float*       __restrict__ outL,
                           unsigned*    __restrict__ cnt8) {
  __shared__ float tile[256 * C_LB];                  // 20480 bytes
  const int t = threadIdx.x;
  const unsigned lbase = (unsigned)(unsigned long long)(uintptr_t)tile;
  const unsigned long long gbase =
      (unsigned long long)(uintptr_t)labels +
      (unsigned long long)blockIdx.x * (256ull * C_LB * 4ull);

  // 5 * 256 * 16B = 20480B == 256 rows * 80B: exact cover, fully coalesced.
#pragma unroll
  for (int i = 0; i < 5; ++i) {
    unsigned off = (unsigned)(i * 256 + t) * 16u;
    async_ld_b128(lbase + off, gbase + off);
  }
  wait_async0();
  __syncthreads();

  const int pid = blockIdx.x * 256 + t;
  const int v   = idx8[pid];
  gatom_add_u32((unsigned long long)(uintptr_t)(cnt8 + v), 1u);
  const unsigned long long dst =
      (unsigned long long)(uintptr_t)(outL + (long long)v * C_LB);
  RowAtomAdd<C_LB>::run(dst, &tile[t * C_LB]);        // 2-way LDS conflict max
}

// ---- points: 1024-point tile, 256 threads, 36 KB LDS ----------------------
__global__ __launch_bounds__(256)
void scatter_points_kernel(const float* __restrict__ points,
                           const int*   __restrict__ idx1,
                           float*       __restrict__ outF,
                           unsigned*    __restrict__ cnt1) {
  __shared__ float tile[1024 * C_PT];                 // 36864 bytes
  const int t = threadIdx.x;
  const unsigned lbase = (unsigned)(unsigned long long)(uintptr_t)tile;
  const unsigned long long gbase =
      (unsigned long long)(uintptr_t)points +
      (unsigned long long)blockIdx.x * (1024ull * C_PT * 4ull);

  // 9 * 256 * 16B = 36864B == 1024 rows * 36B: exact cover, fully coalesced.
#pragma unroll
  for (int i = 0; i < 9; ++i) {
    unsigned off = (unsigned)(i * 256 + t) * 16u;
    async_ld_b128(lbase + off, gbase + off);
  }
  wait_async0();
  __syncthreads();

#pragma unroll
  for (int q = 0; q < 4; ++q) {
    const int pl  = t + q * 256;                      // point within tile
    const int pid = blockIdx.x * 1024 + pl;           // global point id
    const int v   = idx1[pid];
    gatom_add_u32((unsigned long long)(uintptr_t)(cnt1 + v), 1u);
    const unsigned long long dst =
        (unsigned long long)(uintptr_t)(outF + (long long)v * C_PT);
    RowAtomAdd<C_PT>::run(dst, &tile[pl * C_PT]);     // stride-9: conflict-free
  }
}

// ---- normalize: sum / max(count,1), in place ------------------------------
template <int C>
__global__ __launch_bounds__(256)
void normalize_kernel(float* __restrict__ data,
                      const unsigned* __restrict__ cnt,
                      int V) {
  const long long i = (long long)blockIdx.x * 256 + threadIdx.x;
  if (i >= (long long)V * C) return;
  unsigned c = cnt[i / C];                            // constant divisor -> magic mul
  if (c == 0u) c = 1u;
  data[i] = data[i] * (1.0f / (float)c);
}

extern "C" void kernel_launch(void* const* d_in, const int* in_sizes, int n_in,
                              void* d_out, int out_size, void* d_ws, size_t ws_size,
                              hipStream_t stream) {
  const float* points = (const float*)d_in[0];   // [N, 9]  f32
  const int*   idx1   = (const int*)  d_in[1];   // [N]     int
  const float* labels = (const float*)d_in[2];   // [N, 20] f32
  const int*   idx8   = (const int*)  d_in[3];   // [N]     int

  float* outF = (float*)d_out;                        // [V1, 9]
  float* outL = outF + (long long)V1_ * C_PT;         // [V8, 20]

  unsigned* cnt1 = (unsigned*)d_ws;                   // V1 counts
  unsigned* cnt8 = cnt1 + V1_;                        // V8 counts (~2.3 MB total)

  // Zero accumulators every call (graph-capture safe).
  hipMemsetAsync(d_out, 0, (size_t)out_size * sizeof(float), stream);
  hipMemsetAsync(d_ws, 0, (size_t)(V1_ + V8_) * sizeof(unsigned), stream);

  scatter_points_kernel<<<N_PTS / 1024, 256, 0, stream>>>(points, idx1, outF, cnt1);
  scatter_labels_kernel<<<N_PTS / 256, 256, 0, stream>>>(labels, idx8, outL, cnt8);

  const int totF = V1_ * C_PT;                        // 4,500,000
  const int totL = V8_ * C_LB;                        // 1,310,720
  normalize_kernel<C_PT><<<(totF + 255) / 256, 256, 0, stream>>>(outF, cnt1, V1_);
  normalize_kernel<C_LB><<<(totL + 255) / 256, 256, 0, stream>>>(outL, cnt8, V8_);
}